// EmmaGATLayer_15152644620655
// MI455X (gfx1250) — compile-verified
//
#include <hip/hip_runtime.h>

// ---------------------------------------------------------------------------
// GAT layer for MI455X (gfx1250, wave32).
//   xp = x @ W^T            -> fp32 WMMA 16x16x4 (keeps exact fp32 semantics)
//   alphas, segment softmax, scatter-aggregate -> L2-resident atomics
// ---------------------------------------------------------------------------

typedef __attribute__((ext_vector_type(2))) float v2f;
typedef __attribute__((ext_vector_type(8))) float v8f;

#define FDIM 128
#define HEADS 4
#define CHAN 32
#define NEG_SLOPE 0.2f
#define EPS_W 1e-16f

// ---------------- Kernel 1: xp = x @ W^T via V_WMMA_F32_16X16X4_F32 --------
// Block = 256 threads (8 waves). Block owns 16 rows of x, staged once in LDS;
// wave w computes the 16x16 tile for output columns [16w, 16w+16).
__global__ __launch_bounds__(256) void gat_gemm_f32wmma(
    const float* __restrict__ x, const float* __restrict__ W,
    float* __restrict__ xp, int n_nodes) {
  __shared__ float xs[16][132];  // pad 132 -> conflict-free LDS reads

  const int m0 = blockIdx.x * 16;
  const int tid = threadIdx.x;

  // Cooperative load of the 16x128 x-tile (coalesced, 8 floats/thread).
  for (int i = tid; i < 16 * FDIM; i += 256) {
    const int r = i >> 7, c = i & 127;
    const int row = m0 + r;
    xs[r][c] = (row < n_nodes) ? x[(size_t)row * FDIM + c] : 0.0f;
  }
  __syncthreads();

  const int wave = tid >> 5;           // 0..7 : column tile
  const int lane = tid & 31;
  const int l15 = lane & 15;
  const int khalf = (lane >> 4) * 2;   // K phase: 0 (lanes 0-15) or 2 (16-31)
  const int j0 = wave * 16;

  // Lane's W row (output column j0+l15), offset by its K phase.
  const float* __restrict__ wrow = W + (size_t)(j0 + l15) * FDIM + khalf;
  const float* __restrict__ arow = &xs[l15][khalf];

  v8f acc = {};
#pragma unroll 8
  for (int k = 0; k < FDIM; k += 4) {
    // A frag: x[m0+l15, k+khalf .. +1]   (16x4 fp32, ISA layout)
    v2f a = *(const v2f*)(arow + k);
    // B frag: W^T[k+khalf.., j0+l15] = W[j0+l15, k+khalf..]
    v2f b = *(const v2f*)(wrow + k);
    acc = __builtin_amdgcn_wmma_f32_16x16x4_f32(
        /*neg_a=*/false, a, /*neg_b=*/false, b,
        /*c_mod=*/(short)0, acc, /*reuse_a=*/false, /*reuse_b=*/false);
  }

  // C layout: VGPR r -> row = r + 8*(lane>=16), col = l15.
  const int rbase = (lane >> 4) * 8;
  float* __restrict__ outp = xp + (size_t)(m0 + rbase) * FDIM + j0 + l15;
#pragma unroll
  for (int r = 0; r < 8; ++r) {
    if (m0 + rbase + r < n_nodes) outp[(size_t)r * FDIM] = acc[r];
  }
}

// ---------------- Kernel 2: per-node alphas + accumulator init -------------
// One 128-thread block per node; each wave32 = one head, shuffle-reduced.
__global__ __launch_bounds__(128) void gat_alpha_init(
    const float* __restrict__ xp, const float* __restrict__ att_src,
    const float* __restrict__ att_dst, float* __restrict__ src_a,
    float* __restrict__ dst_a, unsigned* __restrict__ maxa,
    float* __restrict__ sumw, float* __restrict__ agg) {
  const int n = blockIdx.x;
  const int j = threadIdx.x;              // j = h*32 + c, flat over [H,C]
  const float v = xp[(size_t)n * FDIM + j];
  float s = v * att_src[j];
  float d = v * att_dst[j];
#pragma unroll
  for (int off = 16; off > 0; off >>= 1) {
    s += __shfl_xor(s, off, 32);
    d += __shfl_xor(d, off, 32);
  }
  if ((j & 31) == 0) {
    const int h = j >> 5;
    src_a[(size_t)n * HEADS + h] = s;
    dst_a[(size_t)n * HEADS + h] = d;
  }
  // Re-initialize accumulators every call (graph replay determinism).
  agg[(size_t)n * FDIM + j] = 0.0f;
  if (j < HEADS) {
    maxa[(size_t)n * HEADS + j] = 0u;  // monotone-mapped "-inf"
    sumw[(size_t)n * HEADS + j] = 0.0f;
  }
}

// Monotone float<->uint map so unsigned atomicMax == float max.
__device__ __forceinline__ unsigned fmap(float x) {
  unsigned u = __float_as_uint(x);
  return (u & 0x80000000u) ? ~u : (u | 0x80000000u);
}
__device__ __forceinline__ float funmap(unsigned u) {
  return __uint_as_float((u & 0x80000000u) ? (u & 0x7FFFFFFFu) : ~u);
}

// ---------------- Kernel 3: per-(edge,head) segment max --------------------
__global__ __launch_bounds__(256) void gat_edge_max(
    const long long* __restrict__ ei, const float* __restrict__ src_a,
    const float* __restrict__ dst_a, unsigned* __restrict__ maxa,
    long long E) {
  const long long t = (long long)blockIdx.x * blockDim.x + threadIdx.x;
  if (t >= E * HEADS) return;
  const long long e = t >> 2;
  const int h = (int)(t & 3);
  const long long dst = ei[e];
  const long long src = ei[E + e];
  float v = src_a[src * HEADS + h] + dst_a[dst * HEADS + h];
  v = (v > 0.0f) ? v : NEG_SLOPE * v;
  atomicMax(&maxa[dst * HEADS + h], fmap(v));
}

// ---------------- Kernel 4: exp + scatter aggregation ----------------------
// 128 threads per edge (one wave per head); 2 edges per 256-thread block.
// agg/xp/sumw all live in the 192MB L2, so atomics stay on-chip.
__global__ __launch_bounds__(256) void gat_edge_agg(
    const long long* __restrict__ ei, const float* __restrict__ xp,
    const float* __restrict__ src_a, const float* __restrict__ dst_a,
    const unsigned* __restrict__ maxa, float* __restrict__ sumw,
    float* __restrict__ agg, long long E) {
  const long long e = (long long)blockIdx.x * 2 + (threadIdx.x >> 7);
  if (e >= E) return;
  const int j = threadIdx.x & 127;
  const int h = j >> 5;
  const int c = j & 31;
  const long long dst = ei[e];
  const long long src = ei[E + e];
  float v = src_a[src * HEADS + h] + dst_a[dst * HEADS + h];
  v = (v > 0.0f) ? v : NEG_SLOPE * v;
  const float w = expf(v - funmap(maxa[dst * HEADS + h]));
  if (c == 0) atomicAdd(&sumw[dst * HEADS + h], w);
  atomicAdd(&agg[dst * FDIM + j], w * xp[src * FDIM + j]);
}

// ---------------- Kernel 5: finalize ---------------------------------------
__global__ __launch_bounds__(256) void gat_finalize(
    const float* __restrict__ xp, const float* __restrict__ sumw,
    const float* __restrict__ bias, float* __restrict__ out, long long total) {
  const long long t = (long long)blockIdx.x * blockDim.x + threadIdx.x;
  if (t >= total) return;
  const int j = (int)(t & 127);
  const long long n = t >> 7;
  const int h = j >> 5;
  out[t] = out[t] / fmaxf(sumw[n * HEADS + h], EPS_W) + xp[t] + bias[j];
}

// ---------------------------------------------------------------------------
extern "C" void kernel_launch(void* const* d_in, const int* in_sizes, int n_in,
                              void* d_out, int out_size, void* d_ws,
                              size_t ws_size, hipStream_t stream) {
  const float* x = (const float*)d_in[0];
  const long long* ei = (const long long*)d_in[1];  // int64 [2,E]: [dst;src]
  const float* W = (const float*)d_in[2];
  const float* att_src = (const float*)d_in[3];
  const float* att_dst = (const float*)d_in[4];
  const float* bias = (const float*)d_in[5];

  const int n_nodes = in_sizes[0] / FDIM;
  const long long E = (long long)in_sizes[1] / 2;
  float* out = (float*)d_out;  // doubles as agg accumulator

  // Workspace carve-up (fp32): xp[N,128] | src_a[N,4] | dst_a[N,4] |
  //                            maxa(u32)[N,4] | sumw[N,4]   (~57.6 MB)
  float* xp = (float*)d_ws;
  float* src_a = xp + (size_t)n_nodes * FDIM;
  float* dst_a = src_a + (size_t)n_nodes * HEADS;
  unsigned* maxa = (unsigned*)(dst_a + (size_t)n_nodes * HEADS);
  float* sumw = (float*)(maxa + (size_t)n_nodes * HEADS);

  // 1) GEMM via fp32 WMMA (N=100000 -> 6250 row tiles)
  gat_gemm_f32wmma<<<(n_nodes + 15) / 16, 256, 0, stream>>>(x, W, xp, n_nodes);

  // 2) alphas + re-init accumulators
  gat_alpha_init<<<n_nodes, 128, 0, stream>>>(xp, att_src, att_dst, src_a,
                                              dst_a, maxa, sumw, out);

  // 3) segment max over dst
  const long long t4 = E * HEADS;
  gat_edge_max<<<(unsigned)((t4 + 255) / 256), 256, 0, stream>>>(ei, src_a,
                                                                 dst_a, maxa, E);

  // 4) exp + scatter-aggregate (L2-resident atomics)
  gat_edge_agg<<<(unsigned)((E + 1) / 2), 256, 0, stream>>>(ei, xp, src_a,
                                                            dst_a, maxa, sumw,
                                                            out, E);

  // 5) normalize + residual + bias
  const long long total = (long long)n_nodes * FDIM;
  gat_finalize<<<(unsigned)((total + 255) / 256), 256, 0, stream>>>(
      xp, sumw, bias, out, total);
}